// Inverse_CDF_Sampling_69372311765557
// MI455X (gfx1250) — compile-verified
//
#include <hip/hip_runtime.h>
#include <hip/hip_bf16.h>

typedef __attribute__((ext_vector_type(2))) float v2f;
typedef __attribute__((ext_vector_type(8))) float v8f;

// ---------------------------------------------------------------------------
// Phase 1: colsum[t] = sum_c feat[0, c, t] via V_WMMA_F32_16X16X4_F32.
// A = ones(16x4)  =>  D[m,n] = sum_k B[k,n] + C[m,n]  (all rows m identical,
// and invariant to the K<->lane mapping of B, so layout ambiguity is harmless).
//
// Unrolled 4x with 4 independent accumulators: 8 loads issued per iteration
// before any s_wait, no WMMA->WMMA RAW chain -> maximal memory-level
// parallelism for this HBM-bound reduction (32 MB @ 23.3 TB/s ~ 1.4 us floor).
// blockIdx.y selects a 128-channel chunk (16 chunks -> 4096 waves in flight).
// Partial sums are written per-chunk (no atomics -> deterministic FP).
// ---------------------------------------------------------------------------
__global__ void icdf_colsum_wmma(const float* __restrict__ feat,
                                 float* __restrict__ partial,
                                 int T, int cChunk) {
    const int lane   = threadIdx.x & 31;
    const int waveId = threadIdx.x >> 5;
    const int t0     = (blockIdx.x * 8 + waveId) * 16;   // 8 waves / block
    const int cBegin = blockIdx.y * cChunk;
    const int cEnd   = cBegin + cChunk;

    const int n  = lane & 15;     // column within tile
    const int kh = lane >> 4;     // which K pair this half-wave holds

    const float* colBase = feat + (size_t)(t0 + n);      // batch 0 only

    v2f a; a.x = 1.0f; a.y = 1.0f;                       // ones A-matrix
    v8f acc0 = {}, acc1 = {}, acc2 = {}, acc3 = {};

    for (int c = cBegin; c < cEnd; c += 16) {
        const size_t r = (size_t)(c + kh * 2) * (size_t)T;
        const size_t sT = (size_t)T;
        // Issue all 8 loads up front (4 independent k-groups of 4 channels).
        v2f b0, b1, b2, b3;
        b0.x = colBase[r];            b0.y = colBase[r + sT];
        b1.x = colBase[r + 4  * sT];  b1.y = colBase[r + 5  * sT];
        b2.x = colBase[r + 8  * sT];  b2.y = colBase[r + 9  * sT];
        b3.x = colBase[r + 12 * sT];  b3.y = colBase[r + 13 * sT];
        // 4 independent WMMAs: no accumulator RAW between consecutive issues.
        acc0 = __builtin_amdgcn_wmma_f32_16x16x4_f32(
            false, a, false, b0, (short)0, acc0, false, false);
        acc1 = __builtin_amdgcn_wmma_f32_16x16x4_f32(
            false, a, false, b1, (short)0, acc1, false, false);
        acc2 = __builtin_amdgcn_wmma_f32_16x16x4_f32(
            false, a, false, b2, (short)0, acc2, false, false);
        acc3 = __builtin_amdgcn_wmma_f32_16x16x4_f32(
            false, a, false, b3, (short)0, acc3, false, false);
    }

    // C/D layout: VGPR0, lanes 0..15 hold M=0, N=lane. Fixed-order merge.
    if (lane < 16) {
        const float s = (acc0[0] + acc1[0]) + (acc2[0] + acc3[0]);
        partial[(size_t)blockIdx.y * (size_t)T + (size_t)(t0 + lane)] = s;
    }
}

// ---------------------------------------------------------------------------
// Phase 2 (single block, 1024 threads): merge partials, normalize, cumsum,
// cdf_int, first-occurrence argmin per target -> idx[t_size].
// ---------------------------------------------------------------------------
__global__ void icdf_build_idx(const float* __restrict__ partial,
                               int* __restrict__ idxOut,
                               int T, int nChunks, int t_size) {
    __shared__ float col[4096];
    __shared__ float chunkPre[1025];   // exclusive prefix of 4-elem chunks; [1024]=total
    __shared__ int   cdfInt[4096];

    const int tid = threadIdx.x;

    // Merge channel-chunk partial sums (deterministic order).
    for (int t = tid; t < T; t += 1024) {
        float s = 0.0f;
        for (int k = 0; k < nChunks; ++k)
            s += partial[(size_t)k * (size_t)T + (size_t)t];
        col[t] = s;
    }
    __syncthreads();

    // Local inclusive scan of 4 consecutive elements per thread.
    const int base = tid * 4;
    float e0 = col[base + 0];
    float e1 = col[base + 1];
    float e2 = col[base + 2];
    float e3 = col[base + 3];
    const float s0 = e0;
    const float s1 = s0 + e1;
    const float s2 = s1 + e2;
    const float s3 = s2 + e3;
    chunkPre[tid] = s3;
    __syncthreads();

    // Serial exclusive scan over 1024 chunk sums (tiny).
    if (tid == 0) {
        float run = 0.0f;
        for (int i = 0; i < 1024; ++i) {
            const float v = chunkPre[i];
            chunkPre[i] = run;
            run += v;
        }
        chunkPre[1024] = run;   // grand total
    }
    __syncthreads();

    const float total = chunkPre[1024];
    const float pre   = chunkPre[tid];
    const float scale = (float)t_size / total;   // cdf * t_size in one step
    const int   hi    = t_size - 1;

    // (cdf * t_size) truncated toward zero, clipped to a_max = t_size-1.
    int v0 = (int)((pre + s0) * scale); if (v0 > hi) v0 = hi;
    int v1 = (int)((pre + s1) * scale); if (v1 > hi) v1 = hi;
    int v2 = (int)((pre + s2) * scale); if (v2 > hi) v2 = hi;
    int v3 = (int)((pre + s3) * scale); if (v3 > hi) v3 = hi;
    cdfInt[base + 0] = v0;
    cdfInt[base + 1] = v1;
    cdfInt[base + 2] = v2;
    cdfInt[base + 3] = v3;
    __syncthreads();

    // First-occurrence argmin of |cdf_int[j] - target| (matches jnp.argmin).
    if (tid < t_size) {
        int best  = 0x7FFFFFFF;
        int bestj = 0;
        for (int j = 0; j < T; ++j) {
            int d = cdfInt[j] - tid;
            d = (d < 0) ? -d : d;
            if (d < best) { best = d; bestj = j; }
        }
        idxOut[tid] = bestj;
    }
}

// ---------------------------------------------------------------------------
// Phase 3: out[row, t'] = feat[row * T + idx[t']], one block per (b,c) row.
// Writes are fully coalesced; reads hit L2 (entire gathered set ~8MB << 192MB).
// ---------------------------------------------------------------------------
__global__ void icdf_gather(const float* __restrict__ feat,
                            const int* __restrict__ idx,
                            float* __restrict__ out,
                            int T, int t_size) {
    const int row = blockIdx.x;
    const int t   = threadIdx.x;
    if (t < t_size) {
        out[(size_t)row * (size_t)t_size + (size_t)t] =
            feat[(size_t)row * (size_t)T + (size_t)idx[t]];
    }
}

extern "C" void kernel_launch(void* const* d_in, const int* in_sizes, int n_in,
                              void* d_out, int out_size, void* d_ws, size_t ws_size,
                              hipStream_t stream) {
    const float* feat = (const float*)d_in[0];

    constexpr int B = 8, C = 2048, T = 4096;
    const int t_size = out_size / (B * C);   // 126 for the reference shapes

    constexpr int nChunks = 16;
    constexpr int cChunk  = C / nChunks;     // 128 channels per chunk

    float* partial = (float*)d_ws;                               // nChunks*T floats
    int*   idx     = (int*)((char*)d_ws + (size_t)nChunks * T * sizeof(float));

    // Phase 1: 8 waves/block * 16 cols/wave = 128 cols per block; 16 c-chunks.
    dim3 g1(T / 128, nChunks);
    icdf_colsum_wmma<<<g1, 256, 0, stream>>>(feat, partial, T, cChunk);

    // Phase 2: single workgroup scan + argmin.
    icdf_build_idx<<<1, 1024, 0, stream>>>(partial, idx, T, nChunks, t_size);

    // Phase 3: gather, one block per (b, c) row.
    int g3block = ((t_size + 63) / 64) * 64;
    if (g3block < 64) g3block = 64;
    icdf_gather<<<B * C, g3block, 0, stream>>>(feat, idx, (float*)d_out, T, t_size);
}